// Block_68633577390412
// MI455X (gfx1250) — compile-verified
//
#include <hip/hip_runtime.h>
#include <hip/hip_bf16.h>

#define T_LEN   65536
#define C_CH    64
#define B_N     4
#define WSZ     512
#define STEP    256
#define NFRAMES 256

typedef __attribute__((ext_vector_type(2))) float v2f;
typedef __attribute__((ext_vector_type(8))) float v8f;

// ---------------------------------------------------------------------------
// Pass 1: y[b,d,t] = sum_c mixer[c,d] * x[b,c,t]   (64x64 GEMM over B*T cols)
// One wave -> 64(M=d) x 16(N=t) strip. A = mixer^T staged in LDS as [c][d].
// V_WMMA_F32_16X16X4_F32: A 16x4 (2 VGPR), B 4x16 (2 VGPR), C/D 16x16 (8 VGPR)
// ---------------------------------------------------------------------------
__global__ __launch_bounds__(256) void mix_wmma_kernel(const float* __restrict__ x,
                                                       const float* __restrict__ mixer,
                                                       float* __restrict__ y) {
    __shared__ float sM[64 * 64];                 // sM[c*64 + d] = mixer[c][d]
    for (int i = threadIdx.x; i < 64 * 64; i += 256) sM[i] = mixer[i];
    __syncthreads();

    const int wave = threadIdx.x >> 5;
    const int lane = threadIdx.x & 31;
    const int half = lane >> 4;                   // 0: lanes 0-15, 1: lanes 16-31
    const int l15  = lane & 15;

    const int tile = blockIdx.x * 8 + wave;       // 16384 tiles total
    const int b    = tile >> 12;                  // / 4096
    const int t0   = (tile & 4095) << 4;          // * 16
    const float* __restrict__ xb = x + (size_t)b * C_CH * T_LEN;

    v8f acc0 = {}, acc1 = {}, acc2 = {}, acc3 = {};

#pragma unroll
    for (int ks = 0; ks < 16; ++ks) {
        const int c0 = ks * 4 + 2 * half;         // K row pair handled by this half-wave
        // B fragment: VGPR0 -> K=c0, VGPR1 -> K=c0+1, N = t0 + l15
        v2f bf;
        bf.x = xb[(size_t)(c0    ) * T_LEN + t0 + l15];
        bf.y = xb[(size_t)(c0 + 1) * T_LEN + t0 + l15];

        // A fragments per m-block: M = mb*16 + l15, K = c0 / c0+1
        v2f a0; a0.x = sM[(c0) * 64 +  0 + l15]; a0.y = sM[(c0 + 1) * 64 +  0 + l15];
        v2f a1; a1.x = sM[(c0) * 64 + 16 + l15]; a1.y = sM[(c0 + 1) * 64 + 16 + l15];
        v2f a2; a2.x = sM[(c0) * 64 + 32 + l15]; a2.y = sM[(c0 + 1) * 64 + 32 + l15];
        v2f a3; a3.x = sM[(c0) * 64 + 48 + l15]; a3.y = sM[(c0 + 1) * 64 + 48 + l15];

        acc0 = __builtin_amdgcn_wmma_f32_16x16x4_f32(false, a0, false, bf, (short)0, acc0, false, false);
        acc1 = __builtin_amdgcn_wmma_f32_16x16x4_f32(false, a1, false, bf, (short)0, acc1, false, false);
        acc2 = __builtin_amdgcn_wmma_f32_16x16x4_f32(false, a2, false, bf, (short)0, acc2, false, false);
        acc3 = __builtin_amdgcn_wmma_f32_16x16x4_f32(false, a3, false, bf, (short)0, acc3, false, false);
    }

    // C/D layout: VGPR r -> M = r + 8*half (within 16-row block), N = l15
    float* __restrict__ yb = y + (size_t)b * C_CH * T_LEN + t0 + l15;
#pragma unroll
    for (int r = 0; r < 8; ++r) {
        const int dd = r + 8 * half;
        yb[(size_t)( 0 + dd) * T_LEN] = acc0[r];
        yb[(size_t)(16 + dd) * T_LEN] = acc1[r];
        yb[(size_t)(32 + dd) * T_LEN] = acc2[r];
        yb[(size_t)(48 + dd) * T_LEN] = acc3[r];
    }
}

// ---------------------------------------------------------------------------
// Pass 2: per (b,c): stream 256 frames: FFT-512 -> freq-domain IIR scan ->
// IFFT-512 -> Hann -> overlap-add -> tanh(gain*s). In-place on io.
// Frame inputs are double-buffered in LDS via GLOBAL_LOAD_ASYNC_TO_LDS_B32
// (ASYNCcnt), so frame f+1's global reads overlap frame f's FFT work.
// Each thread prefetches exactly the slots it later consumes -> only a
// per-wave s_wait_asynccnt is needed for the staging buffer.
// ---------------------------------------------------------------------------
__global__ __launch_bounds__(256) void stream_fft_kernel(const float* __restrict__ gainp,
                                                         const float* __restrict__ transfer,
                                                         float* __restrict__ io) {
    __shared__ float re[WSZ],  im[WSZ];          // FFT work array
    __shared__ float Yre[WSZ], Yim[WSZ];         // scan state
    __shared__ float twr[256], twi[256];         // exp(-i*2pi*k/512), k<256
    __shared__ float trext[WSZ];                 // transfer mirrored to 512 bins
    __shared__ float wind[WSZ];                  // hann
    __shared__ float carry[STEP];                // OLA carry
    __shared__ float stage[2][WSZ];              // async double buffer

    const int tid = threadIdx.x;
    const int bc  = blockIdx.x;                  // 0..255 -> (b,c)
    const int c   = bc & (C_CH - 1);
    float* __restrict__ base = io + (size_t)bc * T_LEN;

    const float PI2 = 6.28318530717958647692f;
    {
        float s, co;
        __sincosf(-PI2 * (float)tid * (1.0f / (float)WSZ), &s, &co);
        twr[tid] = co; twi[tid] = s;
    }
    for (int i = tid; i < WSZ; i += 256) {
        wind[i] = 0.5f * (1.0f - __cosf(PI2 * (float)i * (1.0f / (float)WSZ)));
        const int k = (i <= WSZ / 2) ? i : (WSZ - i);
        trext[i] = transfer[c * (WSZ / 2 + 1) + k];
        Yre[i] = 0.0f; Yim[i] = 0.0f;
    }
    carry[tid] = 0.0f;
    const float gain = gainp[0];
    __syncthreads();

    // Issue async prefetch of frame f1 into stage[f1&1] (clamped addresses;
    // tail zeroing happens at consume time).
    auto issue_prefetch = [&](int f1) {
        if (f1 < NFRAMES) {
            const int buf = f1 & 1;
#pragma unroll
            for (int i = tid; i < WSZ; i += 256) {
                const int t  = f1 * STEP + i;
                const int tc = (t < T_LEN) ? t : (T_LEN - 1);
                const unsigned ldsa = (unsigned)(size_t)&stage[buf][i];
                const unsigned voff = (unsigned)tc * 4u;
                asm volatile("global_load_async_to_lds_b32 %0, %1, %2 offset:0"
                             :: "v"(ldsa), "v"(voff), "s"(base)
                             : "memory");
            }
        }
    };

    issue_prefetch(0);

    for (int f = 0; f < NFRAMES; ++f) {
        // ---- wait for this frame's staged data (per-wave counter) ----
        asm volatile("s_wait_asynccnt 0x0" ::: "memory");
        // ---- kick off next frame's loads into the other buffer ----
        issue_prefetch(f + 1);

        // ---- consume staged frame into bit-reversed positions ----
        {
            const int buf = f & 1;
#pragma unroll
            for (int i = tid; i < WSZ; i += 256) {
                const int t = f * STEP + i;
                const float v = (t < T_LEN) ? stage[buf][i] : 0.0f;
                const int r = (int)(__brev((unsigned)i) >> 23);
                re[r] = v; im[r] = 0.0f;
            }
        }
        __syncthreads();

        // ---- forward FFT-512 (DIT radix-2), 1 butterfly/thread/stage ----
#pragma unroll
        for (int s = 0; s < 9; ++s) {
            const int hs  = 1 << s;
            const int pos = tid & (hs - 1);
            const int i0  = ((tid >> s) << (s + 1)) + pos;
            const int i1  = i0 + hs;
            const int ti  = pos << (8 - s);
            const float wr = twr[ti], wi = twi[ti];
            const float ur = re[i0], ui = im[i0];
            const float vr = re[i1], vi = im[i1];
            const float tr = vr * wr - vi * wi;
            const float tq = vr * wi + vi * wr;
            re[i0] = ur + tr; im[i0] = ui + tq;
            re[i1] = ur - tr; im[i1] = ui - tq;
            __syncthreads();
        }

        // ---- frequency-domain IIR scan: Y = (S + Y) * transfer ----
        for (int i = tid; i < WSZ; i += 256) {
            const float tr = trext[i];
            const float yr = (re[i] + Yre[i]) * tr;
            const float yi = (im[i] + Yim[i]) * tr;
            Yre[i] = yr; Yim[i] = yi;
        }
        __syncthreads();

        // ---- inverse FFT via conj trick: ifft(Y) = conj(fft(conj Y))/N ----
        for (int i = tid; i < WSZ; i += 256) {
            const int r = (int)(__brev((unsigned)i) >> 23);
            re[r] = Yre[i]; im[r] = -Yim[i];
        }
        __syncthreads();
#pragma unroll
        for (int s = 0; s < 9; ++s) {
            const int hs  = 1 << s;
            const int pos = tid & (hs - 1);
            const int i0  = ((tid >> s) << (s + 1)) + pos;
            const int i1  = i0 + hs;
            const int ti  = pos << (8 - s);
            const float wr = twr[ti], wi = twi[ti];
            const float ur = re[i0], ui = im[i0];
            const float vr = re[i1], vi = im[i1];
            const float tr = vr * wr - vi * wi;
            const float tq = vr * wi + vi * wr;
            re[i0] = ur + tr; im[i0] = ui + tq;
            re[i1] = ur - tr; im[i1] = ui - tq;
            __syncthreads();
        }

        // ---- window, overlap-add, tanh, write 256 samples ----
        {
            const float inv = 1.0f / (float)WSZ;
            const float a0 = re[tid]        * inv * wind[tid];
            const float a1 = re[tid + STEP] * inv * wind[tid + STEP];
            const float v  = carry[tid] + a0;
            base[f * STEP + tid] = tanhf(v * gain);
            carry[tid] = a1;          // each thread owns slot tid: no race
        }
        __syncthreads();
    }
}

extern "C" void kernel_launch(void* const* d_in, const int* in_sizes, int n_in,
                              void* d_out, int out_size, void* d_ws, size_t ws_size,
                              hipStream_t stream) {
    const float* x        = (const float*)d_in[0];   // (4,64,65536)
    const float* mixer    = (const float*)d_in[1];   // (64,64)
    const float* transfer = (const float*)d_in[2];   // (64,257)
    const float* gain     = (const float*)d_in[3];   // (1,)
    float* out = (float*)d_out;                      // (4,64,65536)

    // Pass 1: 16384 tiles of 64x16, 8 waves/block -> 2048 blocks
    mix_wmma_kernel<<<2048, 256, 0, stream>>>(x, mixer, out);
    // Pass 2: one workgroup per (b,c), in-place on out
    stream_fft_kernel<<<B_N * C_CH, 256, 0, stream>>>(gain, transfer, out);
}